// Registration_2362232013147
// MI455X (gfx1250) — compile-verified
//
#include <hip/hip_runtime.h>
#include <math.h>
#include <cmath>

typedef __attribute__((ext_vector_type(2))) float v2f;
typedef __attribute__((ext_vector_type(8))) float v8f;

// ---------------------------------------------------------------------------
// Kernel 1: pose composition  T = pose(_rot, _xyz) @ rt_inv
// Writes ws[0..8]  = Rw (row-major 3x3), ws[9..11] = tw.
// One thread; pure function of inputs (deterministic per launch).
// ---------------------------------------------------------------------------
__global__ void pose_setup_kernel(const float* __restrict__ rt_inv,
                                  const float* __restrict__ rot,
                                  const float* __restrict__ xyz,
                                  float* __restrict__ ws) {
    if (threadIdx.x != 0 || blockIdx.x != 0) return;

    const float rx = rot[0], ry = rot[1], rz = rot[2];
    const float th2 = rx * rx + ry * ry + rz * rz;
    const float th  = sqrtf(th2 + 1e-30f);
    float a, b;
    if (th2 < 1e-12f) {              // matches reference's grad-safe branch
        a = 1.0f - th2 / 6.0f;
        b = 0.5f - th2 / 24.0f;
    } else {
        a = sinf(th) / th;
        b = (1.0f - cosf(th)) / th2;
    }

    // K = skew(rot), K2 = K @ K
    float K[3][3] = {{0.0f, -rz,  ry},
                     { rz, 0.0f, -rx},
                     {-ry,  rx, 0.0f}};
    float K2[3][3];
    for (int i = 0; i < 3; ++i)
        for (int j = 0; j < 3; ++j) {
            float s = 0.0f;
            for (int k = 0; k < 3; ++k) s += K[i][k] * K[k][j];
            K2[i][j] = s;
        }

    float R[3][3];
    for (int i = 0; i < 3; ++i)
        for (int j = 0; j < 3; ++j)
            R[i][j] = ((i == j) ? 1.0f : 0.0f) + a * K[i][j] + b * K2[i][j];

    // pose P (4x4)
    float P[4][4];
    for (int i = 0; i < 3; ++i) {
        for (int j = 0; j < 3; ++j) P[i][j] = R[i][j];
        P[i][3] = xyz[i];
    }
    P[3][0] = 0.0f; P[3][1] = 0.0f; P[3][2] = 0.0f; P[3][3] = 1.0f;

    // T = P @ rt_inv   (rt_inv row-major 4x4)
    float T[4][4];
    for (int i = 0; i < 4; ++i)
        for (int j = 0; j < 4; ++j) {
            float s = 0.0f;
            for (int k = 0; k < 4; ++k) s += P[i][k] * rt_inv[k * 4 + j];
            T[i][j] = s;
        }

    for (int i = 0; i < 3; ++i)
        for (int j = 0; j < 3; ++j)
            ws[i * 3 + j] = T[i][j];
    ws[9]  = T[0][3];
    ws[10] = T[1][3];
    ws[11] = T[2][3];
}

// ---------------------------------------------------------------------------
// Kernel 2: ray-cast with trilinear sampling.
//
// Ray generation uses V_WMMA_F32_16X16X4_F32: D(16x16) = A(16x4) x B(4x16).
//   A  = 3x3 matrix in rows 0..2 (K padded to 4 with zeros):
//        lanes 0-15 supply K=0,1 (VGPR0,1); lanes 16-31 supply K=2,3.
//   B  = 16 pixel vectors in columns: lanes 0-15 supply rows K=0,1
//        (u,v or cam_x,cam_y); lanes 16-31 supply rows K=2,3 (z-part, 0).
//   D  : VGPR r holds row M=r, lane n holds column N=n  ->  lane n's
//        d[0..2] are exactly pixel n's x/y/z output. Two batches per wave
//        (pixels in lanes 0-15, then 16-31), one shfl to redistribute.
// WMMA runs before the bounds guard so EXEC is all-ones (ISA requirement).
//
// Hot loop: one thread/pixel, 8x8 blocks -> wave32 = 8x4 pixel tile
// (cacheline reuse; 64MB volume is L2-resident in MI455X's 192MB L2),
// EXEC-masked gather skip outside the cube, software-pipelined with
// global_prefetch_b8 of the next step's voxel rows.
// ---------------------------------------------------------------------------
__global__ __launch_bounds__(64)
void drr_raycast_kernel(const float* __restrict__ vol,
                        const float* __restrict__ kinv,
                        const float* __restrict__ sdd_p,
                        const int* __restrict__ height_p,
                        const int* __restrict__ width_p,
                        const int* __restrict__ nsteps_p,
                        const float* __restrict__ pose,   // ws: Rw(9), tw(3)
                        float* __restrict__ out,
                        int D) {
    const int W = width_p[0];
    const int H = height_p[0];
    const int n = nsteps_p[0];

    const int x = blockIdx.x * 8 + threadIdx.x;
    const int y = blockIdx.y * 8 + threadIdx.y;

    const float sdd = sdd_p[0];
    const float twx = pose[9], twy = pose[10], twz = pose[11];

    // ---------------- WMMA ray generation (EXEC all-ones here) -------------
    const unsigned lane = __lane_id();
    const unsigned ln15 = lane & 15u;
    const bool lo = lane < 16u;

    // A operands: k_inv rows (stage 1) and Rw rows (stage 2), K-padded.
    v2f aK = {0.0f, 0.0f};
    v2f aR = {0.0f, 0.0f};
    if (ln15 < 3u) {
        if (lo) {
            aK.x = kinv[ln15 * 3 + 0]; aK.y = kinv[ln15 * 3 + 1];
            aR.x = pose[ln15 * 3 + 0]; aR.y = pose[ln15 * 3 + 1];
        } else {
            aK.x = kinv[ln15 * 3 + 2]; aK.y = 0.0f;   // K=2 col; K=3 pad = 0
            aR.x = pose[ln15 * 3 + 2]; aR.y = 0.0f;
        }
    }

    const float u = (float)x + 0.5f;
    const float v = (float)y + 0.5f;
    const float u_hi = __shfl(u, (int)(ln15 + 16u), 32);
    const float v_hi = __shfl(v, (int)(ln15 + 16u), 32);

    const v8f zero8 = {};

    // Stage 1: cam = k_inv @ [u,v,1]   (then * sdd)
    v2f b0, b1;
    b0.x = lo ? u    : 1.0f;  b0.y = lo ? v    : 0.0f;   // hi lanes: K=2 row = 1, K=3 = 0
    b1.x = lo ? u_hi : 1.0f;  b1.y = lo ? v_hi : 0.0f;
    v8f cam0 = __builtin_amdgcn_wmma_f32_16x16x4_f32(false, aK, false, b0,
                                                     (short)0, zero8, false, false);
    v8f cam1 = __builtin_amdgcn_wmma_f32_16x16x4_f32(false, aK, false, b1,
                                                     (short)0, zero8, false, false);

    const float c0x = cam0[0] * sdd, c0y = cam0[1] * sdd, c0z = cam0[2] * sdd;
    const float c1x = cam1[0] * sdd, c1y = cam1[1] * sdd, c1z = cam1[2] * sdd;

    // Stage 2: ray = Rw @ cam  (tw cancels in tgt - src)
    const float c0z_b = __shfl(c0z, (int)ln15, 32);   // hi lane 16+n <- lane n's cam_z
    const float c1z_b = __shfl(c1z, (int)ln15, 32);
    v2f d0, d1;
    d0.x = lo ? c0x : c0z_b;  d0.y = lo ? c0y : 0.0f;
    d1.x = lo ? c1x : c1z_b;  d1.y = lo ? c1y : 0.0f;
    v8f ray0 = __builtin_amdgcn_wmma_f32_16x16x4_f32(false, aR, false, d0,
                                                     (short)0, zero8, false, false);
    v8f ray1 = __builtin_amdgcn_wmma_f32_16x16x4_f32(false, aR, false, d1,
                                                     (short)0, zero8, false, false);

    // Redistribute batch-1 results to lanes 16-31 (lane 16+n <- lane n).
    const float r1x = __shfl(ray1[0], (int)ln15, 32);
    const float r1y = __shfl(ray1[1], (int)ln15, 32);
    const float r1z = __shfl(ray1[2], (int)ln15, 32);
    const float rxr = lo ? ray0[0] : r1x;
    const float ryr = lo ? ray0[1] : r1y;
    const float rzr = lo ? ray0[2] : r1z;
    // -----------------------------------------------------------------------

    if (x >= W || y >= H) return;

    const float fn     = (float)n;
    const float inv_n  = 1.0f / fn;
    const float seglen = sqrtf(rxr * rxr + ryr * ryr + rzr * rzr) / fn;

    const float ctr = 0.5f * (float)(D - 1);
    const float bx = twx + ctr, by = twy + ctr, bz = twz + ctr;
    const float Dm1f = (float)(D - 1);
    const int   Dm2  = D - 2;
    const size_t szD = (size_t)D;

    // ---- pipelined state for step s ----
    int   ix = 0, iy = 0, iz = 0;
    float fx = 0.0f, fy = 0.0f, fz = 0.0f;
    bool  inside = false;
    {
        const float alpha = 0.5f * inv_n;
        const float px = fmaf(alpha, rxr, bx);
        const float py = fmaf(alpha, ryr, by);
        const float pz = fmaf(alpha, rzr, bz);
        inside = (px >= 0.0f) && (px <= Dm1f) &&
                 (py >= 0.0f) && (py <= Dm1f) &&
                 (pz >= 0.0f) && (pz <= Dm1f);
        const float wx = floorf(px), wy = floorf(py), wz = floorf(pz);
        fx = px - wx; fy = py - wy; fz = pz - wz;
        int tix = (int)wx, tiy = (int)wy, tiz = (int)wz;
        tix = tix < 0 ? 0 : (tix > Dm2 ? Dm2 : tix);
        tiy = tiy < 0 ? 0 : (tiy > Dm2 ? Dm2 : tiy);
        tiz = tiz < 0 ? 0 : (tiz > Dm2 ? Dm2 : tiz);
        ix = tix; iy = tiy; iz = tiz;
    }

    float acc = 0.0f;
    for (int s = 0; s < n; ++s) {
        // --- compute step s+1 state and prefetch its voxel rows ---
        int   jx = ix, jy = iy, jz = iz;
        float gx = fx, gy = fy, gz = fz;
        bool  nin = false;
        if (s + 1 < n) {
            const float alpha = ((float)(s + 1) + 0.5f) * inv_n;
            const float px = fmaf(alpha, rxr, bx);
            const float py = fmaf(alpha, ryr, by);
            const float pz = fmaf(alpha, rzr, bz);
            nin = (px >= 0.0f) && (px <= Dm1f) &&
                  (py >= 0.0f) && (py <= Dm1f) &&
                  (pz >= 0.0f) && (pz <= Dm1f);
            const float wx = floorf(px), wy = floorf(py), wz = floorf(pz);
            gx = px - wx; gy = py - wy; gz = pz - wz;
            int tix = (int)wx, tiy = (int)wy, tiz = (int)wz;
            tix = tix < 0 ? 0 : (tix > Dm2 ? Dm2 : tix);
            tiy = tiy < 0 ? 0 : (tiy > Dm2 ? Dm2 : tiy);
            tiz = tiz < 0 ? 0 : (tiz > Dm2 ? Dm2 : tiz);
            jx = tix; jy = tiy; jz = tiz;
            if (nin) {
                const float* pp = vol + ((size_t)jx * szD + (size_t)jy) * szD + (size_t)jz;
                __builtin_prefetch(pp, 0, 3);            // (x0,y0,z0..) row
                __builtin_prefetch(pp + szD, 0, 3);      // (x0,y1,z0..) row
            }
        }

        // --- gather + trilerp for step s (skipped entirely when outside) ---
        if (inside) {
            const float* p0 = vol + ((size_t)ix * szD + (size_t)iy) * szD + (size_t)iz;
            const float* p1 = p0 + szD * szD;
            const float c000 = p0[0],       c001 = p0[1];
            const float c010 = p0[szD],     c011 = p0[szD + 1];
            const float c100 = p1[0],       c101 = p1[1];
            const float c110 = p1[szD],     c111 = p1[szD + 1];
            const float l00 = fmaf(fz, c001 - c000, c000);
            const float l01 = fmaf(fz, c011 - c010, c010);
            const float l10 = fmaf(fz, c101 - c100, c100);
            const float l11 = fmaf(fz, c111 - c110, c110);
            const float l0  = fmaf(fy, l01 - l00, l00);
            const float l1  = fmaf(fy, l11 - l10, l10);
            acc += fmaf(fx, l1 - l0, l0);
        }

        ix = jx; iy = jy; iz = jz;
        fx = gx; fy = gy; fz = gz;
        inside = nin;
    }

    out[(size_t)y * (size_t)W + (size_t)x] = acc * seglen;
}

// ---------------------------------------------------------------------------
// Host launcher.
// Input order (setup_inputs): 0 volume, 1 rt_inv, 2 k_inv, 3 sdd, 4 _rot,
// 5 _xyz, 6 height, 7 width, 8 n_steps. Output: (1,1,H,W) float32.
// ---------------------------------------------------------------------------
extern "C" void kernel_launch(void* const* d_in, const int* in_sizes, int n_in,
                              void* d_out, int out_size, void* d_ws, size_t ws_size,
                              hipStream_t stream) {
    const float* vol    = (const float*)d_in[0];
    const float* rt_inv = (const float*)d_in[1];
    const float* k_inv  = (const float*)d_in[2];
    const float* sdd    = (const float*)d_in[3];
    const float* rot    = (const float*)d_in[4];
    const float* xyz    = (const float*)d_in[5];
    const int*   hgt    = (const int*)d_in[6];
    const int*   wid    = (const int*)d_in[7];
    const int*   nst    = (const int*)d_in[8];
    float*       out    = (float*)d_out;
    float*       ws     = (float*)d_ws;

    // Volume is cubic D^3 (setup: 256^3); derive D from the flat element count.
    const int D = (int)llround(cbrt((double)in_sizes[0]));
    // Image is square H==W (setup: 256x256); derive from out_size for grid dims.
    const int S = (int)llround(sqrt((double)out_size));

    pose_setup_kernel<<<1, 1, 0, stream>>>(rt_inv, rot, xyz, ws);

    dim3 block(8, 8);
    dim3 grid((S + 7) / 8, (S + 7) / 8);
    drr_raycast_kernel<<<grid, block, 0, stream>>>(
        vol, k_inv, sdd, hgt, wid, nst, ws, out, D);
}